// GraphAttentionNetwork_30451318129061
// MI455X (gfx1250) — compile-verified
//
#include <hip/hip_runtime.h>
#include <math.h>

typedef float v2f __attribute__((ext_vector_type(2)));
typedef float v8f __attribute__((ext_vector_type(8)));

#define NEG_SLOPE 0.2f

// ---------------------------------------------------------------------------
// Transpose weights: Wt[c][j] = W[j][c]  (W is k x m, Wt is m x k).
// Tiny matrices (<= 512x128); one thread per element.
// ---------------------------------------------------------------------------
__global__ void transpose_w(const float* __restrict__ W, float* __restrict__ Wt,
                            int k, int m) {
    int idx = blockIdx.x * blockDim.x + threadIdx.x;
    if (idx >= k * m) return;
    int c = idx / k;         // output row (column of W)
    int j = idx - c * k;     // output col (row of W)
    Wt[idx] = W[(size_t)j * m + c];
}

// ---------------------------------------------------------------------------
// WMMA f32 GEMM:  Y[n x m] = X[n x k] @ W[k x m] + bias, with W pre-transposed
// (Wt is m x k row-major).  Each wave computes a 32-row slab (two 16-row WMMA
// row-tiles) across NT 16-col tiles using v_wmma_f32_16x16x4_f32.
//
// A-fragment (ISA 7.12.2, 32-bit A 16x4): lane&15 = row, (lane>>4)*2 = first
// of two contiguous K elements -> one b64 load.
// B-fragment from Wt: lane&15 = output col -> row of Wt, same contiguous K
// pair -> one b64 load with constant 16B/step stride (immediate offsets).
// Tail columns (col >= m) are clamped to m-1; garbage accumulator columns are
// never stored, so the K-loop is completely unconditional.
// The loop is software-pipelined one step deep: the WMMAs consume fragments
// loaded in the previous step while the current step's loads are in flight.
// C/D: vgpr i -> M = i (+8 for lanes 16-31), N = lane&15.
// ---------------------------------------------------------------------------
template <int NT>
__global__ void gemm_wmma_f32(const float* __restrict__ X,
                              const float* __restrict__ Wt,
                              const float* __restrict__ bias,
                              float* __restrict__ Y,
                              int n, int k, int m) {
    const int lane = threadIdx.x & 31;
    const int wave = threadIdx.x >> 5;
    const int tile = blockIdx.x * (blockDim.x >> 5) + wave;
    const int r0 = tile * 32;
    if (r0 >= n) return;                 // uniform per wave: EXEC stays full

    const int halfk = lane >> 4;         // 0 or 1
    const int asub  = halfk * 2;         // first K element of this lane's pair
    const int bcol0 = lane & 15;

    int arow0 = r0 + bcol0;
    int arow1 = arow0 + 16;
    arow0 = arow0 < n ? arow0 : (n - 1); // clamped dup rows are never stored
    arow1 = arow1 < n ? arow1 : (n - 1);

    const float* __restrict__ xp0 = X + (size_t)arow0 * k + asub;
    const float* __restrict__ xp1 = X + (size_t)arow1 * k + asub;

    const float* wp[NT];
#pragma unroll
    for (int t = 0; t < NT; ++t) {
        int c = t * 16 + bcol0;
        if (c > m - 1) c = m - 1;        // clamp tail columns (never stored)
        wp[t] = Wt + (size_t)c * k + asub;
    }

    v8f acc0[NT], acc1[NT];
#pragma unroll
    for (int t = 0; t < NT; ++t) { acc0[t] = (v8f)0.0f; acc1[t] = (v8f)0.0f; }

    // prologue: load first fragments
    v2f a0c = *(const v2f*)xp0;
    v2f a1c = *(const v2f*)xp1;
    v2f bc[NT];
#pragma unroll
    for (int t = 0; t < NT; ++t) bc[t] = *(const v2f*)wp[t];

    for (int kk = 4; kk < k; kk += 4) {
        // issue next step's loads before consuming current fragments
        v2f a0n = *(const v2f*)(xp0 + kk);
        v2f a1n = *(const v2f*)(xp1 + kk);
        v2f bn[NT];
#pragma unroll
        for (int t = 0; t < NT; ++t) bn[t] = *(const v2f*)(wp[t] + kk);
#pragma unroll
        for (int t = 0; t < NT; ++t) {
            acc0[t] = __builtin_amdgcn_wmma_f32_16x16x4_f32(
                false, a0c, false, bc[t], (short)0, acc0[t], false, false);
            acc1[t] = __builtin_amdgcn_wmma_f32_16x16x4_f32(
                false, a1c, false, bc[t], (short)0, acc1[t], false, false);
        }
        a0c = a0n; a1c = a1n;
#pragma unroll
        for (int t = 0; t < NT; ++t) bc[t] = bn[t];
    }
    // epilogue: consume last fragments
#pragma unroll
    for (int t = 0; t < NT; ++t) {
        acc0[t] = __builtin_amdgcn_wmma_f32_16x16x4_f32(
            false, a0c, false, bc[t], (short)0, acc0[t], false, false);
        acc1[t] = __builtin_amdgcn_wmma_f32_16x16x4_f32(
            false, a1c, false, bc[t], (short)0, acc1[t], false, false);
    }

    // store both row slabs: vgpr i -> row base + i, col = t*16 + lane&15
    const int mrow0 = r0 + (halfk ? 8 : 0);
#pragma unroll
    for (int t = 0; t < NT; ++t) {
        int col = t * 16 + bcol0;
        if (col >= m) continue;
        float bv = bias[col];
#pragma unroll
        for (int i = 0; i < 8; ++i) {
            int row = mrow0 + i;
            if (row < n)      Y[(size_t)row * m + col]        = acc0[t][i] + bv;
            if (row + 16 < n) Y[(size_t)(row + 16) * m + col] = acc1[t][i] + bv;
        }
    }
}

// ---------------------------------------------------------------------------
// Per-edge attention score: e = a . leaky_relu(xl[src] + xr[dst])
// One wave32 per edge.  H <= 128 and H % 4 == 0 (128 / 40 here): each lane
// owns 4 contiguous features -> b128 gathers; shuffle reduction.
// ---------------------------------------------------------------------------
__global__ void edge_scores(const float* __restrict__ xl,
                            const float* __restrict__ xr,
                            const float* __restrict__ att,
                            const int* __restrict__ src,
                            const int* __restrict__ dst,
                            float* __restrict__ e, int E, int H) {
    const int edge = (int)((blockIdx.x * blockDim.x + threadIdx.x) >> 5);
    const int lane = threadIdx.x & 31;
    if (edge >= E) return;
    const int s = src[edge], d = dst[edge];
    const float* pl = xl + (size_t)s * H;
    const float* pr = xr + (size_t)d * H;
    float sum = 0.0f;
    const int i0 = lane * 4;
    if (i0 + 3 < H) {
        float4 l = *(const float4*)(pl + i0);
        float4 r = *(const float4*)(pr + i0);
        float4 a = *(const float4*)(att + i0);
        float v;
        v = l.x + r.x; v = v > 0.0f ? v : v * NEG_SLOPE; sum += v * a.x;
        v = l.y + r.y; v = v > 0.0f ? v : v * NEG_SLOPE; sum += v * a.y;
        v = l.z + r.z; v = v > 0.0f ? v : v * NEG_SLOPE; sum += v * a.z;
        v = l.w + r.w; v = v > 0.0f ? v : v * NEG_SLOPE; sum += v * a.w;
    }
    for (int i = 128 + lane; i < H; i += 32) {        // generic tail (unused here)
        float v = pl[i] + pr[i];
        v = v > 0.0f ? v : v * NEG_SLOPE;
        sum += v * att[i];
    }
#pragma unroll
    for (int off = 16; off > 0; off >>= 1)
        sum += __shfl_xor(sum, off, 32);
    if (lane == 0) e[edge] = sum;
}

// monotone float<->uint key so atomicMax(u32) implements float max
__device__ __forceinline__ unsigned fkey(float f) {
    unsigned b = __float_as_uint(f);
    return (b & 0x80000000u) ? ~b : (b | 0x80000000u);
}
__device__ __forceinline__ float funkey(unsigned kk) {
    unsigned b = (kk & 0x80000000u) ? (kk ^ 0x80000000u) : ~kk;
    return __uint_as_float(b);
}

__global__ void seg_max(const float* __restrict__ e, const int* __restrict__ dst,
                        unsigned* __restrict__ mkey, int E) {
    int i = blockIdx.x * blockDim.x + threadIdx.x;
    if (i >= E) return;
    atomicMax(&mkey[dst[i]], fkey(e[i]));
}

__global__ void seg_exp(float* __restrict__ e, const int* __restrict__ dst,
                        const unsigned* __restrict__ mkey,
                        float* __restrict__ den, int E) {
    int i = blockIdx.x * blockDim.x + threadIdx.x;
    if (i >= E) return;
    int d = dst[i];
    float ex = expf(e[i] - funkey(mkey[d]));
    e[i] = ex;
    atomicAdd(&den[d], ex);
}

// out[dst] += (ex/den[dst]) * xl[src]; one wave per edge, 4 features/lane
__global__ void aggregate(const float* __restrict__ ex,
                          const float* __restrict__ den,
                          const float* __restrict__ xl,
                          const int* __restrict__ src,
                          const int* __restrict__ dst,
                          float* __restrict__ out, int E, int H) {
    const int edge = (int)((blockIdx.x * blockDim.x + threadIdx.x) >> 5);
    const int lane = threadIdx.x & 31;
    if (edge >= E) return;
    const int s = src[edge], d = dst[edge];
    const float alpha = ex[edge] / den[d];
    const float* pl = xl + (size_t)s * H;
    float* po = out + (size_t)d * H;
    const int i0 = lane * 4;
    if (i0 + 3 < H) {
        float4 l = *(const float4*)(pl + i0);
        atomicAdd(po + i0 + 0, alpha * l.x);
        atomicAdd(po + i0 + 1, alpha * l.y);
        atomicAdd(po + i0 + 2, alpha * l.z);
        atomicAdd(po + i0 + 3, alpha * l.w);
    }
    for (int i = 128 + lane; i < H; i += 32)          // generic tail (unused here)
        atomicAdd(&po[i], alpha * pl[i]);
}

__global__ void bias_act(float* __restrict__ y, const float* __restrict__ bias,
                         int n, int m, int do_elu) {
    size_t i = (size_t)blockIdx.x * blockDim.x + threadIdx.x;
    if (i >= (size_t)n * m) return;
    int col = (int)(i % m);
    float v = y[i] + bias[col];
    if (do_elu) v = v > 0.0f ? v : expm1f(v);
    y[i] = v;
}

__global__ void fill_u32(unsigned* __restrict__ p, unsigned val, size_t count) {
    size_t i = (size_t)blockIdx.x * blockDim.x + threadIdx.x;
    if (i < count) p[i] = val;
}

// one wave per row (C = 40)
__global__ void log_softmax_rows(const float* __restrict__ h,
                                 float* __restrict__ out, int n, int C) {
    const int row = (int)((blockIdx.x * blockDim.x + threadIdx.x) >> 5);
    const int lane = threadIdx.x & 31;
    if (row >= n) return;
    const float* p = h + (size_t)row * C;
    float mx = -INFINITY;
    for (int i = lane; i < C; i += 32) mx = fmaxf(mx, p[i]);
#pragma unroll
    for (int off = 16; off > 0; off >>= 1)
        mx = fmaxf(mx, __shfl_xor(mx, off, 32));
    float s = 0.0f;
    for (int i = lane; i < C; i += 32) s += expf(p[i] - mx);
#pragma unroll
    for (int off = 16; off > 0; off >>= 1)
        s += __shfl_xor(s, off, 32);
    float lse = logf(s) + mx;
    float* po = out + (size_t)row * C;
    for (int i = lane; i < C; i += 32) po[i] = p[i] - lse;
}

// ---------------------------------------------------------------------------
static inline int ceil_div(long long a, long long b) { return (int)((a + b - 1) / b); }

template <int NT>
static void launch_gemm(const float* X, const float* Wt, const float* b, float* Y,
                        int n, int k, int m, hipStream_t stream) {
    int blocks = ceil_div(n, 8 * 32);   // 8 waves/block, 32 rows/wave
    gemm_wmma_f32<NT><<<blocks, 256, 0, stream>>>(X, Wt, b, Y, n, k, m);
}

// transposes W into wt_buf, then runs the WMMA GEMM reading the transposed copy
static void gemm_dispatch(const float* X, const float* W, const float* b, float* Y,
                          float* wt_buf, int n, int k, int m, hipStream_t stream) {
    transpose_w<<<ceil_div((long long)k * m, 256), 256, 0, stream>>>(W, wt_buf, k, m);
    int nt = (m + 15) / 16;
    switch (nt) {
        case 1: launch_gemm<1>(X, wt_buf, b, Y, n, k, m, stream); break;
        case 2: launch_gemm<2>(X, wt_buf, b, Y, n, k, m, stream); break;
        case 3: launch_gemm<3>(X, wt_buf, b, Y, n, k, m, stream); break;
        case 4: launch_gemm<4>(X, wt_buf, b, Y, n, k, m, stream); break;
        case 5: launch_gemm<5>(X, wt_buf, b, Y, n, k, m, stream); break;
        case 6: launch_gemm<6>(X, wt_buf, b, Y, n, k, m, stream); break;
        case 7: launch_gemm<7>(X, wt_buf, b, Y, n, k, m, stream); break;
        default: launch_gemm<8>(X, wt_buf, b, Y, n, k, m, stream); break;
    }
}

extern "C" void kernel_launch(void* const* d_in, const int* in_sizes, int n_in,
                              void* d_out, int out_size, void* d_ws, size_t ws_size,
                              hipStream_t stream) {
    const float* x     = (const float*)d_in[0];
    const int*   ei    = (const int*)  d_in[1];
    const float* W1l   = (const float*)d_in[2];
    const float* b1l   = (const float*)d_in[3];
    const float* W1r   = (const float*)d_in[4];
    const float* b1r   = (const float*)d_in[5];
    const float* a1    = (const float*)d_in[6];
    const float* bias1 = (const float*)d_in[7];
    const float* W2l   = (const float*)d_in[8];
    const float* b2l   = (const float*)d_in[9];
    const float* W2r   = (const float*)d_in[10];
    const float* b2r   = (const float*)d_in[11];
    const float* a2    = (const float*)d_in[12];
    const float* bias2 = (const float*)d_in[13];

    const int H = in_sizes[3];            // 128
    const int C = in_sizes[9];            // 40
    const int F = in_sizes[2] / H;        // 512
    const int N = in_sizes[0] / F;        // 50000
    const int E = in_sizes[1] / 2;        // 850000
    const int* src = ei;
    const int* dst = ei + E;

    // workspace layout
    float*    xl   = (float*)d_ws;
    float*    xr   = xl + (size_t)N * H;
    float*    hbuf = xr + (size_t)N * H;
    float*    ebuf = hbuf + (size_t)N * H;
    unsigned* mkey = (unsigned*)(ebuf + E);
    float*    den  = (float*)(mkey + N);
    float*    wtb  = den + N;             // transposed-weight scratch (F*H floats max)

    float* h2  = (float*)d_out;                 // final layer-2 features [N x C]
    float* lsm = h2 + (size_t)N * C;            // log-softmax output    [N x C]

    const int TPB = 256;
    const int eblk_wave = ceil_div((long long)E * 32, TPB);   // wave-per-edge kernels
    const int eblk_thr  = ceil_div(E, TPB);                   // thread-per-edge kernels
    const int nblk_thr  = ceil_div(N, TPB);

    // ---------------- layer 1 ----------------
    gemm_dispatch(x, W1l, b1l, xl, wtb, N, F, H, stream);
    gemm_dispatch(x, W1r, b1r, xr, wtb, N, F, H, stream);

    edge_scores<<<eblk_wave, TPB, 0, stream>>>(xl, xr, a1, src, dst, ebuf, E, H);

    fill_u32<<<nblk_thr, TPB, 0, stream>>>(mkey, 0u, (size_t)N);
    fill_u32<<<nblk_thr, TPB, 0, stream>>>((unsigned*)den, 0u, (size_t)N);
    fill_u32<<<ceil_div((long long)N * H, TPB), TPB, 0, stream>>>((unsigned*)hbuf, 0u, (size_t)N * H);

    seg_max<<<eblk_thr, TPB, 0, stream>>>(ebuf, dst, mkey, E);
    seg_exp<<<eblk_thr, TPB, 0, stream>>>(ebuf, dst, mkey, den, E);
    aggregate<<<eblk_wave, TPB, 0, stream>>>(ebuf, den, xl, src, dst, hbuf, E, H);
    bias_act<<<ceil_div((long long)N * H, TPB), TPB, 0, stream>>>(hbuf, bias1, N, H, 1);

    // ---------------- layer 2 ----------------
    gemm_dispatch(hbuf, W2l, b2l, xl, wtb, N, H, C, stream);
    gemm_dispatch(hbuf, W2r, b2r, xr, wtb, N, H, C, stream);

    edge_scores<<<eblk_wave, TPB, 0, stream>>>(xl, xr, a2, src, dst, ebuf, E, C);

    fill_u32<<<nblk_thr, TPB, 0, stream>>>(mkey, 0u, (size_t)N);
    fill_u32<<<nblk_thr, TPB, 0, stream>>>((unsigned*)den, 0u, (size_t)N);
    fill_u32<<<ceil_div((long long)N * C, TPB), TPB, 0, stream>>>((unsigned*)h2, 0u, (size_t)N * C);

    seg_max<<<eblk_thr, TPB, 0, stream>>>(ebuf, dst, mkey, E);
    seg_exp<<<eblk_thr, TPB, 0, stream>>>(ebuf, dst, mkey, den, E);
    aggregate<<<eblk_wave, TPB, 0, stream>>>(ebuf, den, xl, src, dst, h2, E, C);
    bias_act<<<ceil_div((long long)N * C, TPB), TPB, 0, stream>>>(h2, bias2, N, C, 0);

    log_softmax_rows<<<ceil_div((long long)N * 32, TPB), TPB, 0, stream>>>(h2, lsm, N, C);
}